// SimpleJoinModel_10230612099474
// MI455X (gfx1250) — compile-verified
//
#include <hip/hip_runtime.h>

#define D_DIM 256          // feature dim of each side
#define ROW_OUT (2 * D_DIM)

// ---- CDNA5 async global<->LDS path detection -------------------------------
#if defined(__has_builtin)
#if __has_builtin(__builtin_amdgcn_global_load_async_to_lds_b128) && \
    __has_builtin(__builtin_amdgcn_global_store_async_from_lds_b128) && \
    __has_builtin(__builtin_amdgcn_s_wait_asynccnt)
#define HAVE_ASYNC_LDS 1
#else
#define HAVE_ASYNC_LDS 0
#endif
#else
#define HAVE_ASYNC_LDS 0
#endif

#if HAVE_ASYNC_LDS
// Probe-verified signature:
//   (v4i* global, __attribute__((address_space(3))) v4i* lds, Imm offset, Imm cpol)
typedef int v4i __attribute__((vector_size(4 * sizeof(int))));
typedef __attribute__((address_space(3))) v4i lds_v4i;
#endif

// ---- Phase 2: gather + concat, one 64-thread block (2 wave32) per row ------
// Each thread moves one 16B chunk of the right row AND one of the left row.
// The right-row DMA is issued BEFORE the key->pos lookup so it is in flight
// while the two dependent scalar loads resolve. Data path is
// global -> LDS (ASYNCcnt-tracked DMA) -> global with no VGPR staging and
// no lane divergence.
__global__ __launch_bounds__(64) void join_concat(
    const float* __restrict__ left,
    const float* __restrict__ right,
    const int* __restrict__ right_on,
    const int* __restrict__ pos,
    float* __restrict__ out, int n_left) {
  const int i = blockIdx.x;
  const int t = threadIdx.x;

  const float4* rsrc = (const float4*)(right + (size_t)i * D_DIM) + t;

#if HAVE_ASYNC_LDS
  __shared__ float4 stage[128];          // 2 KB: [0..63] left row, [64..127] right row
  // right row: independent of the gather index -> issue immediately
  __builtin_amdgcn_global_load_async_to_lds_b128(
      (v4i*)rsrc, (lds_v4i*)&stage[64 + t], 0, 0);
#endif

  // block-uniform key -> inverse-permutation lookup (scalarized by compiler)
  const int col = right_on[0];
  int k = (int)right[(size_t)i * D_DIM + col];
  int j = (k >= 0 && k < n_left) ? pos[k] : 0;   // unmatched -> row 0 (matches argmax-of-false)
  if (j < 0 || j >= n_left) j = 0;               // guard vs. poisoned workspace

  const float4* lsrc = (const float4*)(left + (size_t)j * D_DIM) + t;
  float4* dst = (float4*)(out + (size_t)i * ROW_OUT);

#if HAVE_ASYNC_LDS
  // gathered left row
  __builtin_amdgcn_global_load_async_to_lds_b128(
      (v4i*)lsrc, (lds_v4i*)&stage[t], 0, 0);
  // each lane stores exactly the chunks it loaded -> per-wave wait suffices
  __builtin_amdgcn_s_wait_asynccnt(0);
  __builtin_amdgcn_global_store_async_from_lds_b128(
      (v4i*)(dst + t), (lds_v4i*)&stage[t], 0, 0);
  __builtin_amdgcn_global_store_async_from_lds_b128(
      (v4i*)(dst + 64 + t), (lds_v4i*)&stage[64 + t], 0, 0);
#else
  dst[t]      = *lsrc;   // fallback: plain b128 copies through VGPRs
  dst[64 + t] = *rsrc;
#endif
}

// ---- Phase 1: build inverse permutation pos[key] = row ---------------------
// lkeys are a permutation of 0..n-1 stored as exact float integers, so the
// join reduces to a direct-indexed inverse map (no hash table needed).
__global__ __launch_bounds__(256) void build_inverse(
    const float* __restrict__ left,
    const int* __restrict__ left_on,
    int* __restrict__ pos, int n) {
  int j = blockIdx.x * blockDim.x + threadIdx.x;
  if (j >= n) return;
  int col = left_on[0];
  int k = (int)left[(size_t)j * D_DIM + col];
  if (k >= 0 && k < n) pos[k] = j;
}

extern "C" void kernel_launch(void* const* d_in, const int* in_sizes, int n_in,
                              void* d_out, int out_size, void* d_ws, size_t ws_size,
                              hipStream_t stream) {
  const float* left     = (const float*)d_in[0];
  const float* right    = (const float*)d_in[1];
  const int*   left_on  = (const int*)d_in[2];
  const int*   right_on = (const int*)d_in[3];
  float*       out      = (float*)d_out;

  const int n_left  = in_sizes[0] / D_DIM;
  const int n_right = in_sizes[1] / D_DIM;

  int* pos = (int*)d_ws;   // n_left ints of scratch

  build_inverse<<<(n_left + 255) / 256, 256, 0, stream>>>(left, left_on, pos, n_left);
  join_concat<<<n_right, 64, 0, stream>>>(left, right, right_on, pos, out, n_left);
}